// GCNLinkPredictor_34265249087906
// MI455X (gfx1250) — compile-verified
//
#include <hip/hip_runtime.h>

typedef __attribute__((ext_vector_type(2))) float v2f;
typedef __attribute__((ext_vector_type(8))) float v8f;

constexpr int IN_CH  = 128;
constexpr int HID_CH = 128;
constexpr int OUT_CH = 64;

// ---------------- degree / normalization ----------------
__global__ void k_deg_init(int* __restrict__ cnt, int n) {
  int i = blockIdx.x * blockDim.x + threadIdx.x;
  if (i < n) cnt[i] = 1;  // self loop contributes 1 to every node's degree
}

__global__ void k_deg_count(const int* __restrict__ dst, int* __restrict__ cnt, int e) {
  int i = blockIdx.x * blockDim.x + threadIdx.x;
  if (i < e) atomicAdd(&cnt[dst[i]], 1);
}

__global__ void k_dinv(const int* __restrict__ cnt, float* __restrict__ dinv, int n) {
  int i = blockIdx.x * blockDim.x + threadIdx.x;
  if (i < n) dinv[i] = rsqrtf((float)cnt[i]);
}

// ---------------- GEMM1: H0 = X @ W1  (f32 WMMA, exact precision) ----------------
// One wave per 16x16 tile of the output; 8 waves/block = the 8 column tiles.
__global__ __launch_bounds__(256) void k_gemm1(const float* __restrict__ X,
                                               const float* __restrict__ W,
                                               float* __restrict__ H, int n) {
  const int lane    = threadIdx.x & 31;
  const int wave    = threadIdx.x >> 5;     // 0..7 -> column tile
  const int rowBase = blockIdx.x * 16;
  const int colBase = wave * 16;
  const int l16     = lane & 15;
  const int kHi     = (lane >> 4) << 1;     // lanes 0-15: K=k0,k0+1 ; lanes 16-31: K=k0+2,k0+3
  int row = rowBase + l16; if (row >= n) row = n - 1;   // branchless clamp (EXEC stays full)
  const float* __restrict__ xr = X + (size_t)row * IN_CH;

  v8f c = {};
  #pragma unroll 4
  for (int k0 = 0; k0 < IN_CH; k0 += 4) {
    const int ka = k0 + kHi;
    v2f a = *(const v2f*)(xr + ka);                         // A: 16x4 tile, ISA layout
    v2f b;
    b.x = W[(size_t)ka       * HID_CH + colBase + l16];     // B: 4x16 tile, ISA layout
    b.y = W[(size_t)(ka + 1) * HID_CH + colBase + l16];
    c = __builtin_amdgcn_wmma_f32_16x16x4_f32(false, a, false, b, (short)0, c, false, false);
  }

  // C/D layout: vgpr r -> rows r (lanes 0-15) / r+8 (lanes 16-31)
  const int mOff = (lane >> 4) << 3;
  float* __restrict__ out = H + (size_t)(rowBase + mOff) * HID_CH + colBase + l16;
  if (rowBase + 16 <= n) {          // block-uniform: full tile -> straight-line 8 stores
    #pragma unroll
    for (int r = 0; r < 8; ++r) out[(size_t)r * HID_CH] = c[r];
  } else {                          // tail tile only
    #pragma unroll
    for (int r = 0; r < 8; ++r)
      if (rowBase + mOff + r < n) out[(size_t)r * HID_CH] = c[r];
  }
}

// ---------------- GEMM2: H2 = relu(AGG1 + b1) @ W2  (bias+ReLU fused into A load) ----------------
__global__ __launch_bounds__(128) void k_gemm2(const float* __restrict__ AGG,
                                               const float* __restrict__ B1,
                                               const float* __restrict__ W,
                                               float* __restrict__ H, int n) {
  const int lane    = threadIdx.x & 31;
  const int wave    = threadIdx.x >> 5;     // 0..3 -> column tile of OUT_CH=64
  const int rowBase = blockIdx.x * 16;
  const int colBase = wave * 16;
  const int l16     = lane & 15;
  const int kHi     = (lane >> 4) << 1;
  int row = rowBase + l16; if (row >= n) row = n - 1;
  const float* __restrict__ ar = AGG + (size_t)row * HID_CH;

  v8f c = {};
  #pragma unroll 4
  for (int k0 = 0; k0 < HID_CH; k0 += 4) {
    const int ka = k0 + kHi;
    v2f a;
    a.x = fmaxf(ar[ka]     + B1[ka],     0.0f);
    a.y = fmaxf(ar[ka + 1] + B1[ka + 1], 0.0f);
    v2f b;
    b.x = W[(size_t)ka       * OUT_CH + colBase + l16];
    b.y = W[(size_t)(ka + 1) * OUT_CH + colBase + l16];
    c = __builtin_amdgcn_wmma_f32_16x16x4_f32(false, a, false, b, (short)0, c, false, false);
  }

  const int mOff = (lane >> 4) << 3;
  float* __restrict__ out = H + (size_t)(rowBase + mOff) * OUT_CH + colBase + l16;
  if (rowBase + 16 <= n) {
    #pragma unroll
    for (int r = 0; r < 8; ++r) out[(size_t)r * OUT_CH] = c[r];
  } else {
    #pragma unroll
    for (int r = 0; r < 8; ++r)
      if (rowBase + mOff + r < n) out[(size_t)r * OUT_CH] = c[r];
  }
}

// ---------------- self-loop init: AGG = H * dinv^2 (full overwrite -> deterministic) ----------------
template <int CH4>
__global__ void k_self_init(const float* __restrict__ H, const float* __restrict__ dinv,
                            float* __restrict__ AGG, int n) {
  int idx = blockIdx.x * blockDim.x + threadIdx.x;
  if (idx >= n * CH4) return;
  int node = idx / CH4;           // CH4 is a power of two -> shift
  float di = dinv[node];
  float w  = di * di;
  float4 h = ((const float4*)H)[idx];
  ((float4*)AGG)[idx] = make_float4(h.x * w, h.y * w, h.z * w, h.w * w);
}

// ---------------- edge scatter, 128 channels: wave per edge, 4 ch per lane ----------------
__global__ __launch_bounds__(256) void k_scatter128(const int* __restrict__ src,
                                                    const int* __restrict__ dst,
                                                    const float* __restrict__ dinv,
                                                    const float* __restrict__ H,
                                                    float* __restrict__ AGG, int E) {
  int gid = blockIdx.x * blockDim.x + threadIdx.x;
  // edge id is wave-uniform; readfirstlane -> SGPR index -> scalar (s_load) fetches
  int e = __builtin_amdgcn_readfirstlane(gid >> 5);
  if (e >= E) return;
  int lane = gid & 31;
  int s = src[e], d = dst[e];
  float w = dinv[s] * dinv[d];
  const float4 h = *(const float4*)(H + (size_t)s * 128 + lane * 4);
  float* o = AGG + (size_t)d * 128 + lane * 4;
  unsafeAtomicAdd(o + 0, h.x * w);   // hardware global_atomic_add_f32
  unsafeAtomicAdd(o + 1, h.y * w);
  unsafeAtomicAdd(o + 2, h.z * w);
  unsafeAtomicAdd(o + 3, h.w * w);
}

// ---------------- edge scatter, 64 channels: wave per edge, 2 ch per lane ----------------
__global__ __launch_bounds__(256) void k_scatter64(const int* __restrict__ src,
                                                   const int* __restrict__ dst,
                                                   const float* __restrict__ dinv,
                                                   const float* __restrict__ H,
                                                   float* __restrict__ AGG, int E) {
  int gid = blockIdx.x * blockDim.x + threadIdx.x;
  int e = __builtin_amdgcn_readfirstlane(gid >> 5);
  if (e >= E) return;
  int lane = gid & 31;
  int s = src[e], d = dst[e];
  float w = dinv[s] * dinv[d];
  const float2 h = *(const float2*)(H + (size_t)s * 64 + lane * 2);
  float* o = AGG + (size_t)d * 64 + lane * 2;
  unsafeAtomicAdd(o + 0, h.x * w);
  unsafeAtomicAdd(o + 1, h.y * w);
}

// ---------------- bias add (in place; AGG2 fully rewritten each call beforehand) ----------------
__global__ void k_bias64(float* __restrict__ Z, const float* __restrict__ b, int n) {
  int idx = blockIdx.x * blockDim.x + threadIdx.x;
  if (idx >= n * 64) return;
  Z[idx] += b[idx & 63];
}

// ---------------- decode: wave per candidate edge, shfl_xor wave32 reduction ----------------
__global__ __launch_bounds__(256) void k_decode(const int* __restrict__ ls,
                                                const int* __restrict__ ld,
                                                const float* __restrict__ Z,
                                                float* __restrict__ out, int L) {
  int gid = blockIdx.x * blockDim.x + threadIdx.x;
  int i = __builtin_amdgcn_readfirstlane(gid >> 5);   // label id is wave-uniform
  if (i >= L) return;
  int lane = gid & 31;
  int a = ls[i], b = ld[i];
  float2 za = *(const float2*)(Z + (size_t)a * 64 + lane * 2);
  float2 zb = *(const float2*)(Z + (size_t)b * 64 + lane * 2);
  float p = za.x * zb.x + za.y * zb.y;
  #pragma unroll
  for (int m = 16; m > 0; m >>= 1) p += __shfl_xor(p, m, 32);
  if (lane == 0) out[i] = p;
}

extern "C" void kernel_launch(void* const* d_in, const int* in_sizes, int n_in,
                              void* d_out, int out_size, void* d_ws, size_t ws_size,
                              hipStream_t stream) {
  const float* x   = (const float*)d_in[0];
  const int*   ei  = (const int*)d_in[1];
  const int*   eli = (const int*)d_in[2];
  const float* W1  = (const float*)d_in[3];
  const float* b1  = (const float*)d_in[4];
  const float* W2  = (const float*)d_in[5];
  const float* b2  = (const float*)d_in[6];

  const int n = in_sizes[0] / IN_CH;
  const int E = in_sizes[1] / 2;
  const int L = in_sizes[2] / 2;
  const int* esrc = ei;
  const int* edst = ei + E;
  const int* lsrc = eli;
  const int* ldst = eli + L;

  char*  ws  = (char*)d_ws;
  size_t off = 0;
  auto take = [&](size_t bytes) -> void* {
    void* p = ws + off;
    off += (bytes + 255) & ~(size_t)255;
    return p;
  };
  int*   cnt  = (int*)  take((size_t)n * sizeof(int));
  float* dinv = (float*)take((size_t)n * sizeof(float));
  float* h0   = (float*)take((size_t)n * IN_CH  * sizeof(float));   // X @ W1
  float* agg1 = (float*)take((size_t)n * HID_CH * sizeof(float));   // conv1 aggregation
  float* h2   = (float*)take((size_t)n * OUT_CH * sizeof(float));   // relu(agg1+b1) @ W2
  float* agg2 = (float*)take((size_t)n * OUT_CH * sizeof(float));   // conv2 aggregation -> z
  (void)ws_size; (void)n_in; (void)out_size;

  // degree / D^{-1/2}
  k_deg_init <<<(n + 255) / 256, 256, 0, stream>>>(cnt, n);
  k_deg_count<<<(E + 255) / 256, 256, 0, stream>>>(edst, cnt, E);
  k_dinv     <<<(n + 255) / 256, 256, 0, stream>>>(cnt, dinv, n);

  // conv1
  k_gemm1        <<<(n + 15) / 16, 256, 0, stream>>>(x, W1, h0, n);
  k_self_init<32><<<((size_t)n * 32 + 255) / 256, 256, 0, stream>>>(h0, dinv, agg1, n);
  k_scatter128   <<<((size_t)E * 32 + 255) / 256, 256, 0, stream>>>(esrc, edst, dinv, h0, agg1, E);

  // conv2 (relu + b1 fused into GEMM2 A-operand)
  k_gemm2        <<<(n + 15) / 16, 128, 0, stream>>>(agg1, b1, W2, h2, n);
  k_self_init<16><<<((size_t)n * 16 + 255) / 256, 256, 0, stream>>>(h2, dinv, agg2, n);
  k_scatter64    <<<((size_t)E * 32 + 255) / 256, 256, 0, stream>>>(esrc, edst, dinv, h2, agg2, E);
  k_bias64       <<<((size_t)n * 64 + 255) / 256, 256, 0, stream>>>(agg2, b2, n);

  // decode
  k_decode       <<<((size_t)L * 32 + 255) / 256, 256, 0, stream>>>(lsrc, ldst, agg2, (float*)d_out, L);
}